// AttentionHead_91276644974625
// MI455X (gfx1250) — compile-verified
//
#include <hip/hip_runtime.h>
#include <hip/hip_bf16.h>
#include <stdint.h>

#define NSEQ   8192
#define DMODEL 2048
#define DK     128
#define KT     32     // k-tile for projection GEMM
#define BR     128    // query rows per block (flash)
#define BC     64     // key cols per iteration (flash)

typedef __attribute__((ext_vector_type(2))) float v2f;
typedef __attribute__((ext_vector_type(8))) float v8f;

static __device__ __forceinline__ v8f wmma_f32(v2f a, v2f b, v8f c) {
    return __builtin_amdgcn_wmma_f32_16x16x4_f32(false, a, false, b, (short)0, c, false, false);
}

// ---------------------------------------------------------------------------
// Kernel 1: Q/K/V projection.  out[n,c] = sum_d x[n,d]*W[c,d] + b[c]
// grid = (NSEQ/128, 3), block = 256 (8 waves). Wave w computes rows w*16..w*16+15,
// all 128 output columns (8 accumulators of 16x16 f32). Small LDS footprint ->
// high occupancy hides global latency.
// ---------------------------------------------------------------------------
__global__ __launch_bounds__(256) void qkv_proj_kernel(
    const float* __restrict__ x,
    const float* __restrict__ Wq, const float* __restrict__ bq,
    const float* __restrict__ Wk, const float* __restrict__ bk,
    const float* __restrict__ Wv, const float* __restrict__ bv,
    float* __restrict__ Qo, float* __restrict__ Ko, float* __restrict__ Vo)
{
    __shared__ float xs[128][KT + 1];   // +1 pad: conflict-free column reads
    __shared__ float wsm[128][KT + 1];

    const int tid  = threadIdx.x;
    const int lane = tid & 31;
    const int wave = tid >> 5;
    const int hh   = lane >> 4;   // lane half (K split)
    const int l16  = lane & 15;
    const int mBase = blockIdx.x * 128;

    const float* W; const float* bias; float* O;
    if (blockIdx.y == 0)      { W = Wq; bias = bq; O = Qo; }
    else if (blockIdx.y == 1) { W = Wk; bias = bk; O = Ko; }
    else                      { W = Wv; bias = bv; O = Vo; }

    v8f acc[8];
    #pragma unroll
    for (int t = 0; t < 8; ++t)
        #pragma unroll
        for (int e = 0; e < 8; ++e) acc[t][e] = 0.0f;

    for (int kb = 0; kb < DMODEL; kb += KT) {
        #pragma unroll
        for (int i = 0; i < 4; ++i) {
            const int idx = tid + i * 256;       // 0..1023
            const int row = idx >> 3;            // 8 float4 per row
            const int c4  = (idx & 7) * 4;
            const float4 xv = *reinterpret_cast<const float4*>(&x[(mBase + row) * DMODEL + kb + c4]);
            xs[row][c4 + 0] = xv.x; xs[row][c4 + 1] = xv.y;
            xs[row][c4 + 2] = xv.z; xs[row][c4 + 3] = xv.w;
            const float4 wv = *reinterpret_cast<const float4*>(&W[row * DMODEL + kb + c4]);
            wsm[row][c4 + 0] = wv.x; wsm[row][c4 + 1] = wv.y;
            wsm[row][c4 + 2] = wv.z; wsm[row][c4 + 3] = wv.w;
        }
        __syncthreads();

        const int mOff = wave * 16;
        #pragma unroll
        for (int k0 = 0; k0 < KT; k0 += 4) {
            v2f a;                                    // A 16x4: lane=M, K=2*hh(+1)
            a.x = xs[mOff + l16][k0 + 2 * hh + 0];
            a.y = xs[mOff + l16][k0 + 2 * hh + 1];
            #pragma unroll
            for (int nt = 0; nt < 8; ++nt) {
                v2f b;                                // B 4x16: B[k][n] = W[n][k]
                b.x = wsm[nt * 16 + l16][k0 + 2 * hh + 0];
                b.y = wsm[nt * 16 + l16][k0 + 2 * hh + 1];
                acc[nt] = wmma_f32(a, b, acc[nt]);
            }
        }
        __syncthreads();
    }

    const int mOff = wave * 16;
    #pragma unroll
    for (int nt = 0; nt < 8; ++nt)
        #pragma unroll
        for (int r = 0; r < 8; ++r) {
            const int row = mBase + mOff + r + hh * 8;   // C/D: M=r (lanes<16), M=r+8
            const int col = nt * 16 + l16;
            O[row * DK + col] = acc[nt][r] + bias[col];
        }
}

// ---------------------------------------------------------------------------
// Async global->LDS copy of one 64x128 K tile and one 64x128 V tile.
// Each of the 256 threads issues 8+8 GLOBAL_LOAD_ASYNC_TO_LDS_B128 (16B/lane).
// K tile stored XOR-swizzled for conflict-free column-pattern B-frag reads:
//   pc4 = c4 ^ ((row&15)<<1)
// V tile stored XOR-swizzled for conflict-free row-pattern B-frag reads:
//   pc4 = c4 ^ ((row&7)<<2)
// ---------------------------------------------------------------------------
static __device__ __forceinline__ void issue_async_kv(
    const float* __restrict__ gK, const float* __restrict__ gV,
    uint32_t ldsK, uint32_t ldsV, int tid)
{
    #pragma unroll
    for (int i = 0; i < 8; ++i) {
        const int idx = tid + i * 256;           // 0..2047 (64 rows x 32 float4)
        const int row = idx >> 5;
        const int c4  = idx & 31;
        const uint32_t gbyte = (uint32_t)((row * DK + c4 * 4) * 4);
        const uint32_t kc4   = (uint32_t)(c4 ^ ((row & 15) << 1));
        const uint32_t vc4   = (uint32_t)(c4 ^ ((row & 7) << 2));
        const uint32_t la_k  = ldsK + (uint32_t)(row * DK * 4) + kc4 * 16u;
        const uint32_t la_v  = ldsV + (uint32_t)(row * DK * 4) + vc4 * 16u;
        asm volatile("global_load_async_to_lds_b128 %0, %1, %2"
                     :: "v"(la_k), "v"(gbyte), "s"(gK) : "memory");
        asm volatile("global_load_async_to_lds_b128 %0, %1, %2"
                     :: "v"(la_v), "v"(gbyte), "s"(gV) : "memory");
    }
}

// ---------------------------------------------------------------------------
// Kernel 2: fused flash attention, fp32 WMMA, online softmax, double-buffered
// K/V tiles streamed by the async-to-LDS engine (ASYNCcnt).
// grid = NSEQ/BR, block = 256. Wave w owns 16 query rows; Q frags live in VGPRs.
// ---------------------------------------------------------------------------
__global__ __launch_bounds__(256) void flash_attn_kernel(
    const float* __restrict__ Qm, const float* __restrict__ Km,
    const float* __restrict__ Vm, float* __restrict__ O)
{
    __shared__ float Ks[2][BC * DK];   // 2 x 32 KB, swizzled
    __shared__ float Vs[2][BC * DK];   // 2 x 32 KB, swizzled
    __shared__ float Ps[BR][BC + 1];   // 33 KB, per-wave P exchange (padded)

    const int tid  = threadIdx.x;
    const int lane = tid & 31;
    const int wave = tid >> 5;
    const int hh   = lane >> 4;
    const int l16  = lane & 15;
    const int mBase = blockIdx.x * BR;
    const int mOff  = wave * 16;
    const float inv_scale = 0.08838834764831845f;  // 1/sqrt(128)

    // Pre-load this wave's Q fragments into registers (scaled by 1/sqrt(dk)).
    v2f aq[DK / 4];
    {
        const int qrow = mBase + mOff + l16;
        #pragma unroll
        for (int k0 = 0; k0 < DK; k0 += 4) {
            const float2 qv = *reinterpret_cast<const float2*>(&Qm[qrow * DK + k0 + 2 * hh]);
            aq[k0 >> 2].x = qv.x * inv_scale;
            aq[k0 >> 2].y = qv.y * inv_scale;
        }
    }

    float mstat[8], lstat[8];
    v8f oacc[8];
    #pragma unroll
    for (int r = 0; r < 8; ++r) { mstat[r] = -__builtin_inff(); lstat[r] = 0.0f; }
    #pragma unroll
    for (int t = 0; t < 8; ++t)
        #pragma unroll
        for (int e = 0; e < 8; ++e) oacc[t][e] = 0.0f;

    const uint32_t ldsK0 = (uint32_t)(size_t)&Ks[0][0];
    const uint32_t ldsK1 = (uint32_t)(size_t)&Ks[1][0];
    const uint32_t ldsV0 = (uint32_t)(size_t)&Vs[0][0];
    const uint32_t ldsV1 = (uint32_t)(size_t)&Vs[1][0];

    // Prologue: start streaming tile 0 into buffer 0.
    issue_async_kv(Km, Vm, ldsK0, ldsV0, tid);

    const int NT = NSEQ / BC;   // 128 key tiles
    for (int jt = 0; jt < NT; ++jt) {
        const int buf = jt & 1;

        // Kick off next tile into the other buffer, then wait for current tile
        // (<=16 outstanding == everything older than the 16 just issued).
        if (jt + 1 < NT) {
            const float* gK = Km + (size_t)(jt + 1) * BC * DK;
            const float* gV = Vm + (size_t)(jt + 1) * BC * DK;
            issue_async_kv(gK, gV, buf ? ldsK0 : ldsK1, buf ? ldsV0 : ldsV1, tid);
            asm volatile("s_wait_asynccnt 0x10" ::: "memory");
        } else {
            asm volatile("s_wait_asynccnt 0x0" ::: "memory");
        }
        __syncthreads();   // tile jt visible to all waves

        const float* KsB = &Ks[buf][0];
        const float* VsB = &Vs[buf][0];

        // S = (Q/sqrt(dk)) @ K^T : 16 x 64 per wave, reduction over DK.
        v8f s[4];
        #pragma unroll
        for (int t = 0; t < 4; ++t)
            #pragma unroll
            for (int e = 0; e < 8; ++e) s[t][e] = 0.0f;

        #pragma unroll
        for (int k0 = 0; k0 < DK; k0 += 4) {
            const v2f a = aq[k0 >> 2];
            const int c4k = k0 >> 2;
            #pragma unroll
            for (int nt = 0; nt < 4; ++nt) {
                const int krow = nt * 16 + l16;              // B[k][n] = K[j+n][k]
                const int pc4  = c4k ^ ((krow & 15) << 1);   // un-swizzle
                const float* p = &KsB[krow * DK + pc4 * 4 + 2 * hh];
                v2f b; b.x = p[0]; b.y = p[1];               // one ds_load_b64
                s[nt] = wmma_f32(a, b, s[nt]);
            }
        }

        // Online softmax per row (register r -> row r + 8*hh of this wave).
        #pragma unroll
        for (int r = 0; r < 8; ++r) {
            float mx = s[0][r];
            #pragma unroll
            for (int nt = 1; nt < 4; ++nt) mx = fmaxf(mx, s[nt][r]);
            #pragma unroll
            for (int d = 1; d < 16; d <<= 1) mx = fmaxf(mx, __shfl_xor(mx, d, 32));
            const float mnew = fmaxf(mstat[r], mx);
            const float f    = __expf(mstat[r] - mnew);
            mstat[r] = mnew;
            float sum = 0.0f;
            #pragma unroll
            for (int nt = 0; nt < 4; ++nt) {
                const float p = __expf(s[nt][r] - mnew);
                s[nt][r] = p;
                sum += p;
            }
            #pragma unroll
            for (int d = 1; d < 16; d <<= 1) sum += __shfl_xor(sum, d, 32);
            lstat[r] = lstat[r] * f + sum;
            #pragma unroll
            for (int nt = 0; nt < 8; ++nt) oacc[nt][r] *= f;
        }

        // Stage P through LDS (wave-local; LDS is in-order per wave).
        #pragma unroll
        for (int nt = 0; nt < 4; ++nt)
            #pragma unroll
            for (int r = 0; r < 8; ++r)
                Ps[mOff + r + hh * 8][nt * 16 + l16] = s[nt][r];

        asm volatile("s_wait_dscnt 0x0" ::: "memory");  // wave-local LDS RAW fence

        // oacc += P (16xBC) @ V (BCx128)
        #pragma unroll
        for (int k0 = 0; k0 < BC; k0 += 4) {
            v2f a;
            a.x = Ps[mOff + l16][k0 + 2 * hh + 0];
            a.y = Ps[mOff + l16][k0 + 2 * hh + 1];
            const int r0 = k0 + 2 * hh;
            const int r1 = r0 + 1;
            #pragma unroll
            for (int nt = 0; nt < 8; ++nt) {
                const int col = nt * 16 + l16;               // B[k][n] = V[j+k][n]
                const int c4v = col >> 2, ev = col & 3;
                v2f b;
                b.x = VsB[r0 * DK + ((c4v ^ ((r0 & 7) << 2)) * 4) + ev];
                b.y = VsB[r1 * DK + ((c4v ^ ((r1 & 7) << 2)) * 4) + ev];
                oacc[nt] = wmma_f32(a, b, oacc[nt]);
            }
        }
        __syncthreads();   // all waves done with buf before it is re-streamed
    }

    // Normalize and store.
    #pragma unroll
    for (int nt = 0; nt < 8; ++nt)
        #pragma unroll
        for (int r = 0; r < 8; ++r) {
            const int row = mBase + mOff + r + hh * 8;
            const int col = nt * 16 + l16;
            O[row * DK + col] = oacc[nt][r] / lstat[r];
        }
}

// ---------------------------------------------------------------------------
extern "C" void kernel_launch(void* const* d_in, const int* in_sizes, int n_in,
                              void* d_out, int out_size, void* d_ws, size_t ws_size,
                              hipStream_t stream) {
    (void)in_sizes; (void)n_in; (void)out_size; (void)ws_size;
    const float* x  = (const float*)d_in[0];
    const float* Wq = (const float*)d_in[1];
    const float* bq = (const float*)d_in[2];
    const float* Wk = (const float*)d_in[3];
    const float* bk = (const float*)d_in[4];
    const float* Wv = (const float*)d_in[5];
    const float* bv = (const float*)d_in[6];
    float* out = (float*)d_out;

    float* Q = (float*)d_ws;                 // 3 x 4 MB scratch (Q,K,V)
    float* K = Q + (size_t)NSEQ * DK;
    float* V = K + (size_t)NSEQ * DK;

    dim3 g1(NSEQ / 128, 3);
    qkv_proj_kernel<<<g1, 256, 0, stream>>>(x, Wq, bq, Wk, bk, Wv, bv, Q, K, V);
    flash_attn_kernel<<<NSEQ / BR, 256, 0, stream>>>(Q, K, V, out);
}